// CenterLoss_35433480192479
// MI455X (gfx1250) — compile-verified
//
#include <hip/hip_runtime.h>
#include <hip/hip_bf16.h>

typedef __attribute__((ext_vector_type(2))) float v2f;
typedef __attribute__((ext_vector_type(8))) float v8f;

#define CL_BATCH 4096
#define CL_FEAT  512

// One wave32 per 16-row tile. Computes dist[i] = ||x_i||^2 + ||c_i||^2 - 2 x_i.c_i
// using V_WMMA_F32_16X16X4_F32 for the cross term (diagonal of X * C^T tile).
__global__ __launch_bounds__(32) void center_dist_kernel(
    const float* __restrict__ x,
    const int* __restrict__ labels,
    const float* __restrict__ centers,
    float* __restrict__ dist)
{
    const int tile = blockIdx.x;            // 0..255
    const int m0   = tile * 16;
    const int lane = threadIdx.x;           // 0..31 (wave32)
    const int r    = lane & 15;             // tile row this lane serves
    const int koff = (lane >> 4) * 2;       // half-wave K sub-offset: 0 or 2

    const int row = m0 + r;
    const int lab = labels[row];

    const float* __restrict__ xrow = x + (size_t)row * CL_FEAT;
    const float* __restrict__ crow = centers + (size_t)lab * CL_FEAT;

    v8f acc = {};            // 16x16 f32 accumulator (diagonal is what we need)
    float sx = 0.0f;         // partial ||x||^2 (this half-wave's K residues)
    float sc = 0.0f;         // partial ||c||^2

    #pragma unroll 4
    for (int k = 0; k < CL_FEAT; k += 4) {
        // A-layout: lanes 0-15 hold (K,K+1) of row M=lane; lanes 16-31 hold (K+2,K+3) of M=lane-16
        v2f a = *(const v2f*)(xrow + k + koff);
        // B-layout (mirrored): lanes 0-15 hold (K,K+1) of col N=lane; lanes 16-31 (K+2,K+3) of N=lane-16
        v2f b = *(const v2f*)(crow + k + koff);
        sx += a.x * a.x + a.y * a.y;
        sc += b.x * b.x + b.y * b.y;
        // 8 args: (neg_a, A, neg_b, B, c_mod, C, reuse_a, reuse_b)
        acc = __builtin_amdgcn_wmma_f32_16x16x4_f32(
            false, a, false, b, (short)0, acc, false, false);
    }

    // Fold half-wave partial norms: lane i and lane i+16 both cover row m0+(i&15).
    sx += __shfl_xor(sx, 16, 32);
    sc += __shfl_xor(sc, 16, 32);

    // Diagonal extraction from C/D layout:
    //   rows 0..7  -> lane i      , VGPR i
    //   rows 8..15 -> lane i + 16 , VGPR i - 8   (lanes 24..31 hold M=8+v == N=lane-16)
    const bool active = (lane < 8) || (lane >= 24);
    const int  di     = (lane < 8) ? lane : (lane - 24);   // VGPR index of diagonal elem

    float xc = acc[0];
    #pragma unroll
    for (int j = 1; j < 8; ++j) xc = (di == j) ? acc[j] : xc;

    if (active) {
        const int orow = m0 + ((lane < 8) ? lane : (lane - 16));
        float d = sx + sc - 2.0f * xc;
        d = fminf(fmaxf(d, 1e-12f), 1e12f);
        dist[orow] = d;
    }
}

// Deterministic fixed-shape reduction of 4096 per-row distances -> mean.
__global__ __launch_bounds__(256) void reduce_mean_kernel(
    const float* __restrict__ dist, float* __restrict__ out)
{
    __shared__ float smem[256];
    float s = 0.0f;
    #pragma unroll
    for (int i = 0; i < CL_BATCH / 256; ++i)
        s += dist[threadIdx.x + i * 256];
    smem[threadIdx.x] = s;
    __syncthreads();
    #pragma unroll
    for (int w = 128; w > 0; w >>= 1) {
        if ((int)threadIdx.x < w) smem[threadIdx.x] += smem[threadIdx.x + w];
        __syncthreads();
    }
    if (threadIdx.x == 0) out[0] = smem[0] * (1.0f / (float)CL_BATCH);
}

extern "C" void kernel_launch(void* const* d_in, const int* in_sizes, int n_in,
                              void* d_out, int out_size, void* d_ws, size_t ws_size,
                              hipStream_t stream) {
    const float* x       = (const float*)d_in[0];
    const int*   labels  = (const int*)d_in[1];
    const float* centers = (const float*)d_in[2];
    float*       dist    = (float*)d_ws;      // 4096 floats = 16 KB scratch
    float*       out     = (float*)d_out;

    center_dist_kernel<<<CL_BATCH / 16, 32, 0, stream>>>(x, labels, centers, dist);
    reduce_mean_kernel<<<1, 256, 0, stream>>>(dist, out);
}